// RankingDirectionalLoss_11158325035094
// MI455X (gfx1250) — compile-verified
//
#include <hip/hip_runtime.h>
#include <hip/hip_bf16.h>
#include <stdint.h>

// ---------------------------------------------------------------------------
// RankingDirectionalLoss for MI455X (gfx1250, wave32).
// price MSE + pairwise margin-ranking hinge over all j>i pairs, N=8192.
// Pairwise differences are produced 256-at-a-time with V_WMMA_F32_16X16X4_F32
// (diff matrix is rank-2: r⊗1 - 1⊗r). Column slices are staged to LDS with
// the Tensor Data Mover (tensor_load_to_lds + s_wait_tensorcnt).
// ---------------------------------------------------------------------------

#define NTOT     8192
#define MARGIN_C 0.01f
#define TILE_COLS 1024          // column slice per block (4KB per array in LDS)
#define ROWS_PER_BLOCK 128      // 8 waves * 16 rows
#define GRID_BX  (NTOT / ROWS_PER_BLOCK)   // 64
#define GRID_BY  (NTOT / TILE_COLS)        // 8
#define NPAIR_PART (GRID_BX * GRID_BY)     // 512
#define NPRICE_PART 32

typedef float v2f  __attribute__((ext_vector_type(2)));
typedef float v8f  __attribute__((ext_vector_type(8)));
typedef unsigned int u32x4 __attribute__((ext_vector_type(4)));
typedef int i32x8 __attribute__((ext_vector_type(8)));
typedef int i32x4 __attribute__((ext_vector_type(4)));

// Low 32 bits of a generic pointer to LDS are the LDS byte offset
// (shared aperture lives entirely in bits 63:32 — see ISA aperture check).
__device__ __forceinline__ unsigned lds_byte_addr(const void* p) {
  return (unsigned)(unsigned long long)(uintptr_t)p;
}

// 1-D contiguous TDM load: nElem f32 from global `src` into LDS at lds_addr.
// Descriptor packing per CDNA5 ISA §8.3/8.4 (group0: count=1,type=2;
// group1: data_size=4B, tensor_dim0=tile_dim0=nElem, stride=nElem; 2D-max
// so groups 2/3 (and the trailing group) are zero).
__device__ __forceinline__ void tdm_load_1d_f32(const float* src,
                                                unsigned lds_addr,
                                                int nElem) {
  unsigned long long ga = (unsigned long long)(uintptr_t)src;
  u32x4 g0;
  g0[0] = 1u;                                            // count=1 (valid), user desc
  g0[1] = lds_addr;                                      // lds_addr [63:32]
  g0[2] = (unsigned)(ga & 0xFFFFFFFFull);                // global_addr[31:0]
  g0[3] = (unsigned)((ga >> 32) & 0x01FFFFFFull)         // global_addr[56:32]
        | (2u << 30);                                    // type=2 ("image")
  unsigned ne = (unsigned)nElem;
  i32x8 g1;
  g1[0] = (int)(2u << 16);                               // wg_mask=0, data_size=2 (4B)
  g1[1] = (int)((ne & 0xFFFFu) << 16);                   // tensor_dim0[15:0]
  g1[2] = (int)(((ne >> 16) & 0xFFFFu) | (1u << 16));    // tensor_dim0[31:16], tensor_dim1=1
  g1[3] = (int)((ne & 0xFFFFu) << 16);                   // tile_dim0 = nElem
  g1[4] = 0;                                             // tile_dim1=0, tile_dim2=0 (1-D tile)
  g1[5] = (int)ne;                                       // tensor_dim0_stride[31:0]
  g1[6] = 0;                                             // stride hi / dim1_stride lo
  g1[7] = 0;
  i32x4 gz4 = {0, 0, 0, 0};
  i32x8 gz8 = {0, 0, 0, 0, 0, 0, 0, 0};
  __builtin_amdgcn_tensor_load_to_lds(g0, g1, gz4, gz4, gz8, 0);
}

// ---------------------------------------------------------------------------
// Kernel A: returns r = (x - prev)/prev and per-block MSE partials.
// ---------------------------------------------------------------------------
__global__ void prep_kernel(const float* __restrict__ pred,
                            const float* __restrict__ act,
                            const float* __restrict__ prev,
                            float* __restrict__ rp,
                            float* __restrict__ ra,
                            float* __restrict__ pricePart) {
  int i = blockIdx.x * 256 + threadIdx.x;
  float p  = pred[i];
  float a  = act[i];
  float pv = prev[i];
  rp[i] = (p - pv) / pv;           // IEEE div: matches JAX rounding exactly
  ra[i] = (a - pv) / pv;
  float d = p - a;

  __shared__ float red[256];
  red[threadIdx.x] = d * d;
  __syncthreads();
  for (int s = 128; s > 0; s >>= 1) {
    if (threadIdx.x < s) red[threadIdx.x] += red[threadIdx.x + s];
    __syncthreads();
  }
  if (threadIdx.x == 0) pricePart[blockIdx.x] = red[0];
}

// ---------------------------------------------------------------------------
// Kernel B: pairwise hinge. Block (bx,by) covers rows [bx*128,+128) x cols
// [by*1024,+1024). Each wave owns a 16-row strip; per 16-col tile two
// V_WMMA_F32_16X16X4_F32 produce the 16x16 pred_diff and act_diff tiles.
// ---------------------------------------------------------------------------
__global__ void pair_kernel(const float* __restrict__ rp,
                            const float* __restrict__ ra,
                            float* __restrict__ pairSum,
                            unsigned* __restrict__ pairCnt) {
  const int bx = blockIdx.x;
  const int by = blockIdx.y;
  const int cs = by * TILE_COLS;

  __shared__ float s_p[TILE_COLS];
  __shared__ float s_a[TILE_COLS];
  __shared__ float redS[8];
  __shared__ unsigned redC[8];

  // Wave 0 stages this block's column slice into LDS via the TDM.
  if (threadIdx.x < 32) {
    tdm_load_1d_f32(rp + cs, lds_byte_addr(s_p), TILE_COLS);
    tdm_load_1d_f32(ra + cs, lds_byte_addr(s_a), TILE_COLS);
    __builtin_amdgcn_s_wait_tensorcnt(0);
  }
  __syncthreads();

  const int lane = threadIdx.x & 31;
  const int wave = threadIdx.x >> 5;
  const int iw   = bx * ROWS_PER_BLOCK + wave * 16;   // first row of this wave
  const bool lo  = lane < 16;
  const int mhalf = lo ? 0 : 8;       // C/D layout: lane>=16 holds M = vgpr+8
  const int ncol  = lane & 15;        // C/D layout: N = lane % 16

  // A (16x4 f32): lanes 0-15 hold {K=0,K=1} = {r_i, 1}; lanes 16-31 {K=2,K=3}=0.
  float rpi = lo ? rp[iw + lane] : 0.0f;
  float rai = lo ? ra[iw + lane] : 0.0f;
  const float aone = lo ? 1.0f : 0.0f;
  v2f aP = { rpi, aone };
  v2f aA = { rai, aone };
  const float b0 = lo ? 1.0f : 0.0f;  // B row K=0 = ones (lanes 0-15)
  const v8f zero8 = { 0, 0, 0, 0, 0, 0, 0, 0 };

  float sum = 0.0f;
  int cnt = 0;

  int jstart = (iw > cs) ? iw : cs;   // both multiples of 16; enforce j>i blocks
  int jend = cs + TILE_COLS;
  for (int j0 = jstart; j0 < jend; j0 += 16) {
    // B (4x16 f32): row K=1 = -r_j (lanes 0-15); rows K=2,3 zero.
    float bp1 = lo ? -s_p[j0 - cs + lane] : 0.0f;
    float ba1 = lo ? -s_a[j0 - cs + lane] : 0.0f;
    v2f bP = { b0, bp1 };
    v2f bA = { b0, ba1 };

    // D[i][j] = r_i*1 + 1*(-r_j) = r_i - r_j  (exact: one product + one add, RNE)
    v8f pd = __builtin_amdgcn_wmma_f32_16x16x4_f32(
        false, aP, false, bP, (short)0, zero8, false, false);
    v8f ad = __builtin_amdgcn_wmma_f32_16x16x4_f32(
        false, aA, false, bA, (short)0, zero8, false, false);

    const int jg = j0 + ncol;
#pragma unroll
    for (int e = 0; e < 8; ++e) {
      float adv = ad[e];
      float pdv = pd[e];
      int ig = iw + e + mhalf;
      bool valid = (jg > ig) && (adv != 0.0f);
      float sgn = (adv > 0.0f) ? 1.0f : -1.0f;
      float h = fmaxf(fmaf(-sgn, pdv, MARGIN_C), 0.0f);
      sum += valid ? h : 0.0f;
      cnt += valid ? 1 : 0;
    }
  }

  // wave32 reduction
  for (int off = 16; off > 0; off >>= 1) {
    sum += __shfl_down(sum, off, 32);
    cnt += __shfl_down(cnt, off, 32);
  }
  if (lane == 0) { redS[wave] = sum; redC[wave] = (unsigned)cnt; }
  __syncthreads();
  if (threadIdx.x == 0) {
    float S = 0.0f;
    unsigned C = 0;
    for (int w = 0; w < 8; ++w) { S += redS[w]; C += redC[w]; }
    int slot = bx * gridDim.y + by;
    pairSum[slot] = S;
    pairCnt[slot] = C;
  }
}

// ---------------------------------------------------------------------------
// Kernel C: deterministic fixed-order final reduction + scalar combine.
// ---------------------------------------------------------------------------
__global__ void final_kernel(const float* __restrict__ pairSum,
                             const unsigned* __restrict__ pairCnt,
                             const float* __restrict__ pricePart,
                             float* __restrict__ out) {
  __shared__ float sS[256];
  __shared__ unsigned sC[256];
  __shared__ float sP[256];
  float s = 0.0f; unsigned c = 0; float p = 0.0f;
  for (int i = threadIdx.x; i < NPAIR_PART; i += 256) {
    s += pairSum[i];
    c += pairCnt[i];
  }
  for (int i = threadIdx.x; i < NPRICE_PART; i += 256) p += pricePart[i];
  sS[threadIdx.x] = s; sC[threadIdx.x] = c; sP[threadIdx.x] = p;
  __syncthreads();
  for (int st = 128; st > 0; st >>= 1) {
    if (threadIdx.x < st) {
      sS[threadIdx.x] += sS[threadIdx.x + st];
      sC[threadIdx.x] += sC[threadIdx.x + st];
      sP[threadIdx.x] += sP[threadIdx.x + st];
    }
    __syncthreads();
  }
  if (threadIdx.x == 0) {
    float mse  = sP[0] / (float)NTOT;
    float rank = (sC[0] > 0) ? (sS[0] / (float)sC[0]) : 0.0f;
    out[0] = 0.5f * mse + 0.5f * rank;
  }
}

// ---------------------------------------------------------------------------
extern "C" void kernel_launch(void* const* d_in, const int* in_sizes, int n_in,
                              void* d_out, int out_size, void* d_ws, size_t ws_size,
                              hipStream_t stream) {
  const float* pred = (const float*)d_in[0];
  const float* act  = (const float*)d_in[1];
  const float* prev = (const float*)d_in[2];

  float* ws = (float*)d_ws;
  float* rp        = ws;                        // 8192 f32
  float* ra        = ws + NTOT;                 // 8192 f32
  float* pairSum   = ws + 2 * NTOT;             // 512 f32
  unsigned* pairCnt = (unsigned*)(ws + 2 * NTOT + NPAIR_PART);  // 512 u32
  float* pricePart = ws + 2 * NTOT + 2 * NPAIR_PART;            // 32 f32

  prep_kernel<<<NTOT / 256, 256, 0, stream>>>(pred, act, prev, rp, ra, pricePart);
  dim3 gridB(GRID_BX, GRID_BY);
  pair_kernel<<<gridB, 256, 0, stream>>>(rp, ra, pairSum, pairCnt);
  final_kernel<<<1, 256, 0, stream>>>(pairSum, pairCnt, pricePart, (float*)d_out);
}